// BachNetInferenceWithBeamSearch_20607253086242
// MI455X (gfx1250) — compile-verified
//
#include <hip/hip_runtime.h>

// ---------------- problem constants ----------------
#define PART   62
#define EXTRA  7
#define CR     32
#define HID    4096
#define NC     48                       // num candidates (beam width)
#define IN_SZ  ((5*CR + 1)*PART + (2*CR + 1)*EXTRA)   // 10437
#define IN_A   (IN_SZ + PART)                          // 10499
#define IN_T   (IN_SZ + 2*PART)                        // 10561
#define SPLITK 8

typedef __attribute__((ext_vector_type(2))) float v2f;
typedef __attribute__((ext_vector_type(8))) float v8f;

__device__ __forceinline__ v8f wmma_f32_4(v2f a, v2f b, v8f c) {
    return __builtin_amdgcn_wmma_f32_16x16x4_f32(false, a, false, b, (short)0, c,
                                                 false, false);
}

// =====================================================================
// Split-K GEMM with V_WMMA_F32_16X16X4_F32.
//   partial[s][Mpad][Npad] = A[M x K] * W[K x Nreal]   (W row-major, ld = Nreal)
// One wave owns (s, n-tile) and computes MT m-tiles so the bandwidth-
// critical B (weight) tile is loaded once per k-step and reused MT times.
// - tile/split decomposition scalarized via readfirstlane (SALU loop control)
// - W is streamed with non-temporal loads (read exactly once; keep L2 for
//   the heavily re-read activations and the split-K partial round-trip)
// - K-loop unrolled 4x: all loads issued, then all WMMAs
// =====================================================================
template<int MT, bool MASK_M, bool MASK_N>
__global__ void __launch_bounds__(256)
wmma_gemm_splitk(const float* __restrict__ A, const float* __restrict__ W,
                 float* __restrict__ partial,
                 int M, int K, int Nreal, int Npad, int chunkK)
{
    const int nTiles = Npad >> 4;
    // global wave id: identical in all lanes -> force into an SGPR
    const int gw = __builtin_amdgcn_readfirstlane(
        (blockIdx.x * blockDim.x + threadIdx.x) >> 5);
    if (gw >= SPLITK * nTiles) return;                             // wave-uniform

    const int s    = gw / nTiles;
    const int nt   = gw % nTiles;
    const int lane = threadIdx.x & 31;
    const int half = lane >> 4;        // 0: K rows 0/1,  1: K rows 2/3
    const int l16  = lane & 15;

    const int   ncol  = nt << 4;
    const int   nIdx  = ncol + l16;
    const float bmask = (!MASK_N || nIdx < Nreal) ? 1.f : 0.f;
    const int   nIdxC = (!MASK_N || nIdx < Nreal) ? nIdx : (Nreal - 1);

    const int kstart = s * chunkK;
    const int kend   = (kstart + chunkK < K) ? (kstart + chunkK) : K;

    v8f acc[MT];
    const v8f vzero = {0.f,0.f,0.f,0.f,0.f,0.f,0.f,0.f};
#pragma unroll
    for (int mt = 0; mt < MT; ++mt) acc[mt] = vzero;

    // A row pointers + (optional) row-validity masks per m-tile
    const float* ap[MT];
    float amask[MT];
#pragma unroll
    for (int mt = 0; mt < MT; ++mt) {
        const int r = mt * 16 + l16;
        amask[mt] = (!MASK_M || r < M) ? 1.f : 0.f;
        ap[mt] = A + (size_t)((!MASK_M || r < M) ? r : 0) * (size_t)K
                   + kstart + 2 * half;
    }

    const size_t ldw = (size_t)Nreal;
    const float* wp  = W + (size_t)(kstart + 2*half) * ldw + nIdxC;

    const int span = kend - kstart;
    int k = kstart;

    // ---- main loop: 4 k-steps (16 K) per iteration --------------------
    const int kfast4 = kstart + (span & ~15);
    for (; k < kfast4; k += 16) {
        float bx[4], by[4];
#pragma unroll
        for (int j = 0; j < 4; ++j) {
            bx[j] = __builtin_nontemporal_load(wp + (size_t)(4*j)     * ldw);
            by[j] = __builtin_nontemporal_load(wp + (size_t)(4*j + 1) * ldw);
        }
        __builtin_prefetch(wp + 64 * ldw, 0, 1);   // cover strided row walk
        wp += 16 * ldw;

        float ax[4][MT], ay[4][MT];
#pragma unroll
        for (int j = 0; j < 4; ++j)
#pragma unroll
            for (int mt = 0; mt < MT; ++mt) {
                ax[j][mt] = ap[mt][4*j];
                ay[j][mt] = ap[mt][4*j + 1];
            }
#pragma unroll
        for (int mt = 0; mt < MT; ++mt) ap[mt] += 16;

#pragma unroll
        for (int j = 0; j < 4; ++j) {
            v2f b;
            b.x = MASK_N ? bx[j] * bmask : bx[j];
            b.y = MASK_N ? by[j] * bmask : by[j];
#pragma unroll
            for (int mt = 0; mt < MT; ++mt) {
                v2f a;
                a.x = MASK_M ? ax[j][mt] * amask[mt] : ax[j][mt];
                a.y = MASK_M ? ay[j][mt] * amask[mt] : ay[j][mt];
                acc[mt] = wmma_f32_4(a, b, acc[mt]);
            }
        }
    }

    // ---- leftover full 4-K steps --------------------------------------
    const int kfast = kstart + (span & ~3);
    for (; k < kfast; k += 4) {
        v2f b;
        b.x = __builtin_nontemporal_load(wp)       * (MASK_N ? bmask : 1.f);
        b.y = __builtin_nontemporal_load(wp + ldw) * (MASK_N ? bmask : 1.f);
        wp += 4 * ldw;
#pragma unroll
        for (int mt = 0; mt < MT; ++mt) {
            v2f a;
            a.x = ap[mt][0] * (MASK_M ? amask[mt] : 1.f);
            a.y = ap[mt][1] * (MASK_M ? amask[mt] : 1.f);
            ap[mt] += 4;
            acc[mt] = wmma_f32_4(a, b, acc[mt]);
        }
    }

    // ---- masked K tail (1..3 valid k's) -------------------------------
    if (k < kend) {
        const int k0 = k + 2*half;
        const float m0 = (k0     < kend) ? 1.f : 0.f;
        const float m1 = (k0 + 1 < kend) ? 1.f : 0.f;
        const int k0c = (k0     < kend) ? k0     : (kend - 1);
        const int k1c = (k0 + 1 < kend) ? k0 + 1 : (kend - 1);
        v2f b;
        b.x = __builtin_nontemporal_load(W + (size_t)k0c * ldw + nIdxC) * bmask * m0;
        b.y = __builtin_nontemporal_load(W + (size_t)k1c * ldw + nIdxC) * bmask * m1;
#pragma unroll
        for (int mt = 0; mt < MT; ++mt) {
            const int r  = mt * 16 + l16;
            const int rc = (!MASK_M || r < M) ? r : 0;
            const float am = MASK_M ? amask[mt] : 1.f;
            v2f a;
            a.x = A[(size_t)rc * K + k0c] * (am * m0);
            a.y = A[(size_t)rc * K + k1c] * (am * m1);
            acc[mt] = wmma_f32_4(a, b, acc[mt]);
        }
    }

    // ---- store partial tile:  row = mt*16 + r + 8*half, col = l16 -----
    const int Mpad = MT * 16;
    float* pout = partial + (size_t)s * (size_t)Mpad * (size_t)Npad;
#pragma unroll
    for (int mt = 0; mt < MT; ++mt) {
        const int rowBase = mt * 16 + 8 * half;
#pragma unroll
        for (int r = 0; r < 8; ++r)
            pout[(size_t)(rowBase + r) * Npad + ncol + l16] = acc[mt][r];
    }
}

// ---------------- split-K reduce + bias + optional SELU ----------------
__global__ void reduce_bias_act(const float* __restrict__ partial,
                                const float* __restrict__ bias,
                                float* __restrict__ out,
                                int M, int Nreal, int Npad, int Mpad, int applySelu)
{
    const int i = blockIdx.x * blockDim.x + threadIdx.x;
    if (i >= M * Nreal) return;
    const int m = i / Nreal, n = i % Nreal;
    float v = bias[n];
    const size_t base   = (size_t)m * Npad + n;
    const size_t stride = (size_t)Mpad * Npad;
    for (int s = 0; s < SPLITK; ++s) v += partial[base + s * stride];
    if (applySelu) {
        const float scale = 1.0507009873554805f, alpha = 1.6732632423543772f;
        v = (v > 0.f) ? scale * v : scale * alpha * expm1f(v);
    }
    out[(size_t)m * Npad + n] = v;
}

// ---------------- bass: log_softmax(62) + top-48 ----------------
__global__ void bass_post(const float* __restrict__ logits,
                          float* __restrict__ lp0, int* __restrict__ p_bass)
{
    __shared__ float lg[PART];
    __shared__ float red[64];
    const int t = threadIdx.x;
    if (t < PART) lg[t] = logits[t];
    __syncthreads();
    float m = (t < PART) ? lg[t] : -1e30f;
    red[t] = m; __syncthreads();
    for (int s = 32; s > 0; s >>= 1) { if (t < s) red[t] = fmaxf(red[t], red[t+s]); __syncthreads(); }
    const float mx = red[0]; __syncthreads();
    red[t] = (t < PART) ? expf(lg[t] - mx) : 0.f; __syncthreads();
    for (int s = 32; s > 0; s >>= 1) { if (t < s) red[t] += red[t+s]; __syncthreads(); }
    const float lse = mx + logf(red[0]); __syncthreads();
    if (t < PART) lg[t] -= lse;
    __syncthreads();
    if (t == 0) {                               // top-48 of 62, lowest-index ties
        for (int i = 0; i < NC; ++i) {
            float bv = -1e30f; int bi = 0;
            for (int j = 0; j < PART; ++j) if (lg[j] > bv) { bv = lg[j]; bi = j; }
            lp0[i] = bv; p_bass[i] = bi; lg[bi] = -1e30f;
        }
    }
}

// ---------------- beam expand posts (log_softmax rows + global top-48) ----
__device__ __forceinline__ void beam_select(const float* logits, const float* lp_prev,
                                            float* tot, float* rv, int* ri,
                                            float* sel_v, int* sel_i)
{
    const int t = threadIdx.x;
    if (t < NC) {
        const float* row = logits + t * 64;
        float m = -1e30f;
        for (int j = 0; j < PART; ++j) m = fmaxf(m, row[j]);
        float s = 0.f;
        for (int j = 0; j < PART; ++j) s += expf(row[j] - m);
        const float lse = m + logf(s);
        const float lp  = lp_prev[t];
        for (int j = 0; j < PART; ++j) tot[t * PART + j] = row[j] - lse + lp;
    }
    __syncthreads();
    for (int i = 0; i < NC; ++i) {
        float bv = -1e30f; int bi = 0x7fffffff;
        for (int j = t; j < NC * PART; j += 256) {
            const float v = tot[j];
            if (v > bv || (v == bv && j < bi)) { bv = v; bi = j; }
        }
        rv[t] = bv; ri[t] = bi; __syncthreads();
        for (int s = 128; s > 0; s >>= 1) {
            if (t < s) {
                if (rv[t+s] > rv[t] || (rv[t+s] == rv[t] && ri[t+s] < ri[t])) {
                    rv[t] = rv[t+s]; ri[t] = ri[t+s];
                }
            }
            __syncthreads();
        }
        if (t == 0) { sel_v[i] = rv[0]; sel_i[i] = ri[0]; tot[ri[0]] = -1e30f; }
        __syncthreads();
    }
}

__global__ void alto_post(const float* __restrict__ logits, const float* __restrict__ lp0,
                          const int* __restrict__ p_bass,
                          float* __restrict__ lp1, int* __restrict__ p_alto,
                          int* __restrict__ p_bass2)
{
    __shared__ float tot[NC * PART];
    __shared__ float rv[256];
    __shared__ int   ri[256];
    __shared__ float sv[NC];
    __shared__ int   si[NC];
    beam_select(logits, lp0, tot, rv, ri, sv, si);
    const int t = threadIdx.x;
    if (t < NC) {
        const int idx = si[t];
        lp1[t]     = sv[t];
        p_alto[t]  = idx % PART;
        p_bass2[t] = p_bass[idx / PART];
    }
}

__global__ void tenor_post(const float* __restrict__ logits, const float* __restrict__ lp1,
                           const int* __restrict__ p_bass2, const int* __restrict__ p_alto,
                           float* __restrict__ out)
{
    __shared__ float tot[NC * PART];
    __shared__ float rv[256];
    __shared__ int   ri[256];
    __shared__ float sv[NC];
    __shared__ int   si[NC];
    beam_select(logits, lp1, tot, rv, ri, sv, si);
    const int t = threadIdx.x;
    if (t < NC) {
        const int idx = si[t];
        const int h = idx / PART, p = idx % PART;
        out[t*4 + 0] = sv[t];
        out[t*4 + 1] = (float)p_bass2[h];
        out[t*4 + 2] = (float)p_alto[h];
        out[t*4 + 3] = (float)p;
    }
}

// ---------------- beam input builders ----------------
__global__ void build_xa(const float* __restrict__ x, const int* __restrict__ pb,
                         float* __restrict__ xa)
{
    const int i = blockIdx.x * 256 + threadIdx.x;
    if (i >= NC * IN_A) return;
    const int c = i / IN_A, k = i % IN_A;
    xa[i] = (k < IN_SZ) ? x[k] : ((k - IN_SZ) == pb[c] ? 1.f : 0.f);
}

__global__ void build_xt(const float* __restrict__ x, const int* __restrict__ pb2,
                         const int* __restrict__ pa, float* __restrict__ xt)
{
    const int i = blockIdx.x * 256 + threadIdx.x;
    if (i >= NC * IN_T) return;
    const int c = i / IN_T, k = i % IN_T;
    float v;
    if (k < IN_SZ)             v = x[k];
    else if (k < IN_SZ + PART) v = ((k - IN_SZ)        == pb2[c]) ? 1.f : 0.f;
    else                       v = ((k - IN_SZ - PART) == pa[c])  ? 1.f : 0.f;
    xt[i] = v;
}

// =====================================================================
extern "C" void kernel_launch(void* const* d_in, const int* in_sizes, int n_in,
                              void* d_out, int out_size, void* d_ws, size_t ws_size,
                              hipStream_t stream)
{
    const float* x = (const float*)d_in[0];
    const float *bw1=(const float*)d_in[1],  *bb1=(const float*)d_in[2],
                *bw2=(const float*)d_in[3],  *bb2=(const float*)d_in[4],
                *bw3=(const float*)d_in[5],  *bb3=(const float*)d_in[6];
    const float *aw1=(const float*)d_in[7],  *ab1=(const float*)d_in[8],
                *aw2=(const float*)d_in[9],  *ab2=(const float*)d_in[10],
                *aw3=(const float*)d_in[11], *ab3=(const float*)d_in[12];
    const float *tw1=(const float*)d_in[13], *tb1=(const float*)d_in[14],
                *tw2=(const float*)d_in[15], *tb2=(const float*)d_in[16],
                *tw3=(const float*)d_in[17], *tb3=(const float*)d_in[18];

    // workspace carve-up (floats, 16-elem aligned)
    float* wsf = (float*)d_ws;
    size_t o = 0;
    auto carve = [&](size_t n) { float* p = wsf + o; o += (n + 15) & ~(size_t)15; return p; };
    float* xa      = carve((size_t)NC * IN_A);         // 503,952
    float* xt      = carve((size_t)NC * IN_T);         // 506,928
    float* h1      = carve((size_t)NC * HID);          // 196,608
    float* h2      = carve((size_t)NC * HID);
    float* logitsv = carve((size_t)NC * 64);
    float* partial = carve((size_t)SPLITK * NC * HID); // 1,572,864
    float* lp0     = carve(NC);
    float* lp1     = carve(NC);
    int*   p_bass  = (int*)carve(NC);
    int*   p_alto  = (int*)carve(NC);
    int*   p_bass2 = (int*)carve(NC);
    (void)ws_size; (void)n_in; (void)in_sizes; (void)out_size;

    auto gemm = [&](const float* A, const float* W, int M, int K,
                    int Nreal, int Npad, int mTiles) {
        const int chunkK = (((K + SPLITK - 1) / SPLITK) + 3) & ~3;
        const int waves  = SPLITK * (Npad / 16);
        const int blocks = (waves * 32 + 255) / 256;
        const bool maskN = (Nreal != Npad);
        if (mTiles == 1) {       // bass: M = 1 -> rows masked
            if (maskN)
                wmma_gemm_splitk<1,true,true ><<<blocks,256,0,stream>>>(A,W,partial,M,K,Nreal,Npad,chunkK);
            else
                wmma_gemm_splitk<1,true,false><<<blocks,256,0,stream>>>(A,W,partial,M,K,Nreal,Npad,chunkK);
        } else {                 // alto/tenor: M = 48 = 3*16 -> no row mask
            if (maskN)
                wmma_gemm_splitk<3,false,true ><<<blocks,256,0,stream>>>(A,W,partial,M,K,Nreal,Npad,chunkK);
            else
                wmma_gemm_splitk<3,false,false><<<blocks,256,0,stream>>>(A,W,partial,M,K,Nreal,Npad,chunkK);
        }
    };
    auto reduce = [&](const float* bias, float* outp, int M, int Nreal, int Npad,
                      int mTiles, int selu) {
        const int total = M * Nreal;
        reduce_bias_act<<<(total + 255) / 256, 256, 0, stream>>>(
            partial, bias, outp, M, Nreal, Npad, mTiles * 16, selu);
    };

    // ---- bass (M = 1) ----
    gemm(x,  bw1, 1, IN_SZ, HID, HID, 1);  reduce(bb1, h1, 1, HID, HID, 1, 1);
    gemm(h1, bw2, 1, HID,   HID, HID, 1);  reduce(bb2, h2, 1, HID, HID, 1, 1);
    gemm(h2, bw3, 1, HID,   PART, 64, 1);  reduce(bb3, logitsv, 1, PART, 64, 1, 0);
    bass_post<<<1, 64, 0, stream>>>(logitsv, lp0, p_bass);

    // ---- alto (M = 48) ----
    build_xa<<<(NC * IN_A + 255) / 256, 256, 0, stream>>>(x, p_bass, xa);
    gemm(xa, aw1, NC, IN_A, HID, HID, 3);  reduce(ab1, h1, NC, HID, HID, 3, 1);
    gemm(h1, aw2, NC, HID,  HID, HID, 3);  reduce(ab2, h2, NC, HID, HID, 3, 1);
    gemm(h2, aw3, NC, HID,  PART, 64, 3);  reduce(ab3, logitsv, NC, PART, 64, 3, 0);
    alto_post<<<1, 256, 0, stream>>>(logitsv, lp0, p_bass, lp1, p_alto, p_bass2);

    // ---- tenor (M = 48) ----
    build_xt<<<(NC * IN_T + 255) / 256, 256, 0, stream>>>(x, p_bass2, p_alto, xt);
    gemm(xt, tw1, NC, IN_T, HID, HID, 3);  reduce(tb1, h1, NC, HID, HID, 3, 1);
    gemm(h1, tw2, NC, HID,  HID, HID, 3);  reduce(tb2, h2, NC, HID, HID, 3, 1);
    gemm(h2, tw3, NC, HID,  PART, 64, 3);  reduce(tb3, logitsv, NC, PART, 64, 3, 0);
    tenor_post<<<1, 256, 0, stream>>>(logitsv, lp1, p_bass2, p_alto, (float*)d_out);
}